// GRUDecoder_71683004170542
// MI455X (gfx1250) — compile-verified
//
#include <hip/hip_runtime.h>

typedef __attribute__((ext_vector_type(2))) float v2f;
typedef __attribute__((ext_vector_type(8))) float v8f;

#define D_NEURAL 302
#define D_IN     306
#define GATES    192   // 3 * HIDDEN
#define HIDDEN   64
#define KOUT     4
#define GTILES   (GATES / 16)   // 12

// ---------------------------------------------------------------------------
// Phase 1: GN[t, g] = sum_{c<302} neural[t, c] * W_ih[g, c]
// One wave32 computes a full 16(t) x 192(g) row-block: 12 f32 accumulators,
// one A-fragment load feeds 12 v_wmma_f32_16x16x4_f32 per K-step.
// All fragment loads are unconditional global_load_b64 (8B-aligned);
// the K=300..301 tail is a single peeled WMMA with cndmask-zeroed fragments.
// ---------------------------------------------------------------------------
__global__ __launch_bounds__(128)
void gn_gemm_wmma(const float* __restrict__ neural,
                  const float* __restrict__ W_ih,
                  float* __restrict__ GN,
                  int T, int tiles_t)
{
    const int wave = (blockIdx.x * blockDim.x + threadIdx.x) >> 5;
    if (wave >= tiles_t) return;                 // wave-uniform exit

    const int lane = threadIdx.x & 31;
    const int half = lane >> 4;                  // 0 -> K {0,1}; 1 -> K {2,3}
    const int l    = lane & 15;
    const int t0   = wave << 4;

    int rowA = t0 + l;                           // time index (A row)
    if (rowA >= T) rowA = T - 1;                 // clamp: memory safety only;
                                                 // clamped rows are never stored
    // Lane-half base pointers (both 8-byte aligned for every k0 % 4 == 0).
    const float* arow = neural + (size_t)rowA * D_NEURAL + (half << 1);
    const float* brow = W_ih   + (size_t)l    * D_IN     + (half << 1);

    v8f acc[GTILES];
    #pragma unroll
    for (int g = 0; g < GTILES; ++g) acc[g] = (v8f){};

    // Main K loop: K = 0..299 (75 iterations), fully in-bounds.
    #pragma unroll 3
    for (int k0 = 0; k0 < 300; k0 += 4) {
        const v2f a = *(const v2f*)(arow + k0);  // A[t0+l, k0+2*half .. +1]
        #pragma unroll
        for (int g = 0; g < GTILES; ++g) {
            // B[K, N=g*16+l] = W_ih[g*16+l, K]; same addr reg, imm offset g*19584B
            const v2f b = *(const v2f*)(brow + k0 + g * 16 * D_IN);
            acc[g] = __builtin_amdgcn_wmma_f32_16x16x4_f32(
                false, a, false, b, (short)0, acc[g], false, false);
        }
    }

    // Tail: K = 300,301 live in lane-half 0; lane-half 1 contributes zeros.
    {
        const int kt = half ? 0 : 300;           // safe address for all lanes
        v2f a = *(const v2f*)(arow - (half << 1) + kt);
        if (half) { a.x = 0.0f; a.y = 0.0f; }    // pure v_cndmask, no branch
        #pragma unroll
        for (int g = 0; g < GTILES; ++g) {
            v2f b = *(const v2f*)(brow - (half << 1) + kt + g * 16 * D_IN);
            if (half) { b.x = 0.0f; b.y = 0.0f; }
            acc[g] = __builtin_amdgcn_wmma_f32_16x16x4_f32(
                false, a, false, b, (short)0, acc[g], false, false);
        }
    }

    // D 16x16 f32 layout: VGPR r -> (M=r, N=lane) lanes 0-15,
    //                               (M=r+8, N=lane-16) lanes 16-31.
    #pragma unroll
    for (int g = 0; g < GTILES; ++g) {
        const int colD = (g << 4) + l;
        #pragma unroll
        for (int r = 0; r < 8; ++r) {
            const int rowD = t0 + (half << 3) + r;
            if (rowD < T) GN[(size_t)rowD * GATES + colD] = acc[g][r];
        }
    }
}

// ---------------------------------------------------------------------------
// Phase 2: sequential GRU scan. One block, 192 threads (6 x wave32).
// Thread j owns gate row j: W_hh row in VGPRs, 4 feedback weights in VGPRs.
// h, gates, b_prev live in LDS; GN row streamed from global with prefetch.
// ---------------------------------------------------------------------------
__device__ __forceinline__ float sigmoidf_(float x) {
    return 1.0f / (1.0f + expf(-x));
}

__global__ __launch_bounds__(GATES)
void gru_seq(const float* __restrict__ GN,
             const float* __restrict__ b_gt,
             const float* __restrict__ W_ih,
             const float* __restrict__ W_hh,
             const float* __restrict__ b_ih,
             const float* __restrict__ b_hh,
             const float* __restrict__ W_out,
             const float* __restrict__ b_out,
             const int*   __restrict__ warmup_p,
             float* __restrict__ out,
             int T)
{
    __shared__ __align__(16) float h_lds[HIDDEN];
    __shared__ float r_lds[HIDDEN];
    __shared__ float z_lds[HIDDEN];
    __shared__ float in_lds[HIDDEN];
    __shared__ float hn_lds[HIDDEN];
    __shared__ __align__(16) float wout_lds[KOUT * HIDDEN];
    __shared__ float bout_lds[KOUT];
    __shared__ float bprev[KOUT];

    const int j  = threadIdx.x;           // 0..191
    const int wu = *warmup_p;

    // Preload per-thread weights into VGPRs.
    float whh[HIDDEN];
    #pragma unroll
    for (int k = 0; k < HIDDEN; ++k) whh[k] = W_hh[(size_t)j * HIDDEN + k];
    const float wb0 = W_ih[(size_t)j * D_IN + D_NEURAL + 0];
    const float wb1 = W_ih[(size_t)j * D_IN + D_NEURAL + 1];
    const float wb2 = W_ih[(size_t)j * D_IN + D_NEURAL + 2];
    const float wb3 = W_ih[(size_t)j * D_IN + D_NEURAL + 3];
    const float bi  = b_ih[j];
    const float bh  = b_hh[j];

    // Shared-state init.
    if (j < HIDDEN) h_lds[j] = 0.0f;
    for (int i = j; i < KOUT * HIDDEN; i += GATES) wout_lds[i] = W_out[i];
    if (j < KOUT) { bout_lds[j] = b_out[j]; bprev[j] = b_gt[j]; }  // b_gt[0][:]
    __syncthreads();

    const float4* h4 = (const float4*)h_lds;

    for (int t = 0; t < T; ++t) {
        // Stream this step's precomputed input-gate row; prefetch the next one.
        const float gn = GN[(size_t)t * GATES + j];
        if (t + 1 < T)
            __builtin_prefetch(&GN[(size_t)(t + 1) * GATES + j], 0, 1);

        // gi_j = GN + b_ih + W_ih[:,302:306] . b_prev
        const float gi = gn + bi + wb0 * bprev[0] + wb1 * bprev[1]
                                 + wb2 * bprev[2] + wb3 * bprev[3];
        // gh_j = b_hh + W_hh[j,:] . h   (16 x ds_load_b128)
        float gh = bh;
        #pragma unroll
        for (int k4 = 0; k4 < HIDDEN / 4; ++k4) {
            const float4 hv = h4[k4];
            gh += whh[4 * k4 + 0] * hv.x + whh[4 * k4 + 1] * hv.y
                + whh[4 * k4 + 2] * hv.z + whh[4 * k4 + 3] * hv.w;
        }

        if (j < HIDDEN) {                 // r gate
            r_lds[j] = sigmoidf_(gi + gh);
        } else if (j < 2 * HIDDEN) {      // z gate
            z_lds[j - HIDDEN] = sigmoidf_(gi + gh);
        } else {                          // n gate: keep i_n / h_n split (r * h_n)
            in_lds[j - 2 * HIDDEN] = gi;
            hn_lds[j - 2 * HIDDEN] = gh;
        }
        __syncthreads();

        if (j < HIDDEN) {
            const float n = tanhf(in_lds[j] + r_lds[j] * hn_lds[j]);
            const float z = z_lds[j];
            h_lds[j]      = (1.0f - z) * n + z * h_lds[j];
        }
        __syncthreads();

        if (j < KOUT) {                   // readout + feedback
            float acc = bout_lds[j];
            const float4* w4 = (const float4*)&wout_lds[j * HIDDEN];
            #pragma unroll
            for (int k4 = 0; k4 < HIDDEN / 4; ++k4) {
                const float4 wv = w4[k4];
                const float4 hv = h4[k4];
                acc += wv.x * hv.x + wv.y * hv.y + wv.z * hv.z + wv.w * hv.w;
            }
            out[(size_t)t * KOUT + j] = acc;
            bprev[j] = (t < wu) ? b_gt[(size_t)t * KOUT + j] : acc;
        }
        __syncthreads();
    }
}

// ---------------------------------------------------------------------------
extern "C" void kernel_launch(void* const* d_in, const int* in_sizes, int n_in,
                              void* d_out, int out_size, void* d_ws, size_t ws_size,
                              hipStream_t stream) {
    const float* neural = (const float*)d_in[0];
    const float* b_gt   = (const float*)d_in[1];
    const float* W_ih   = (const float*)d_in[2];
    const float* W_hh   = (const float*)d_in[3];
    const float* b_ih   = (const float*)d_in[4];
    const float* b_hh   = (const float*)d_in[5];
    const float* W_out  = (const float*)d_in[6];
    const float* b_out  = (const float*)d_in[7];
    const int*   warmup = (const int*)d_in[8];
    float*       out    = (float*)d_out;

    const int T = in_sizes[0] / D_NEURAL;          // 65536
    float* GN = (float*)d_ws;                      // T x 192 f32 scratch

    // Phase 1: parallel WMMA GEMM — one wave per 16-row time block.
    const int tiles_t = (T + 15) / 16;             // = #waves
    const int threads = 128;                       // 4 waves / block
    const int wpb     = threads / 32;
    const int blocks  = (tiles_t + wpb - 1) / wpb;
    gn_gemm_wmma<<<blocks, threads, 0, stream>>>(neural, W_ih, GN, T, tiles_t);

    // Phase 2: sequential recurrence, single workgroup.
    gru_seq<<<1, GATES, 0, stream>>>(GN, b_gt, W_ih, W_hh, b_ih, b_hh,
                                     W_out, b_out, warmup, out, T);
}